// MultiSpectralFFTLayerr_87067577025396
// MI455X (gfx1250) — compile-verified
//
#include <hip/hip_runtime.h>

typedef float v2f __attribute__((ext_vector_type(2)));
typedef float v8f __attribute__((ext_vector_type(8)));

#define HEIGHT  56
#define WIDTH   56
#define CHANNEL 512
#define BATCH   32
#define SPATIAL (HEIGHT * WIDTH)        // 3136
#define ROWS    (BATCH * CHANNEL)       // 16384
#define CENTER  (28 * WIDTH + 28)       // 1596 : fftshifted DC bin
#define WAVES_PER_BLOCK 8
#define K_PER_WAVE (SPATIAL / WAVES_PER_BLOCK)  // 392

// ---------------------------------------------------------------------------
// Kernel 1: gather mask[c, 28, 28] into a dense 512-float table in d_ws.
// out[n,c] = mask[c,28,28] * sum_{h,w} x[n,c,h,w]  (exact reduction of the
// fft2 -> shift -> mask -> unshift -> ifft2 -> real -> spatial-sum pipeline).
// ---------------------------------------------------------------------------
__global__ void gather_mask_center(const float* __restrict__ mask,
                                   float* __restrict__ mctr) {
    int c = blockIdx.x * blockDim.x + threadIdx.x;
    if (c < CHANNEL) mctr[c] = mask[(size_t)c * SPATIAL + CENTER];
}

// ---------------------------------------------------------------------------
// Kernel 2: WMMA-driven row-sum. 1024 blocks x 256 threads (8 waves).
// Each block owns 16 consecutive (n,c) rows; each wave owns a 392-element
// K-slice of those rows and reduces it with V_WMMA_F32_16X16X4_F32
// (B = ones -> D[M][N] = rowsum_M). Cross-wave reduce via LDS, then wave 0
// scales by mask-center and stores 16 outputs.
// ---------------------------------------------------------------------------
__global__ void __launch_bounds__(256)
fft_mask_rowsum(const float* __restrict__ x,
                const float* __restrict__ mctr,
                float* __restrict__ out) {
    __shared__ v8f red[WAVES_PER_BLOCK * 32];   // 8 KB

    const int lane = threadIdx.x & 31;
    const int wave = threadIdx.x >> 5;
    const int hi   = lane >> 4;          // 0: lanes 0-15 (K=0,1)  1: lanes 16-31 (K=2,3)
    const int r    = lane & 15;          // row within 16-row tile (A-matrix M = r)
    const int rowBase = blockIdx.x * 16; // tile of rows; 512 % 16 == 0 so never crosses n

    // A-matrix f32 16x4 native layout: this lane supplies row r, K = 2*hi, 2*hi+1
    const float* p = x + (size_t)(rowBase + r) * SPATIAL + wave * K_PER_WAVE + 2 * hi;

    v8f acc = {};                        // C/D accumulator (16x16 f32, 8 VGPRs)
    v2f b;  b.x = 1.0f; b.y = 1.0f;      // B = all ones (layout-independent)

    #pragma unroll 2
    for (int k = 0; k < K_PER_WAVE / 4; ++k) {   // 98 iterations, K=4 each
        v2f a = *(const v2f*)p;                  // global_load_b64, 16B/row stream
        p += 4;
        // D = A x ones + C  ->  D[M][N] = C[M][N] + sum_{K=0..3} x[row M][kwin+K]
        acc = __builtin_amdgcn_wmma_f32_16x16x4_f32(
            /*neg_a=*/false, a, /*neg_b=*/false, b,
            /*c_mod=*/(short)0, acc, /*reuse_a=*/false, /*reuse_b=*/false);
    }

    // Cross-wave reduction of the 8 partial D tiles (lane-wise add keeps layout).
    red[wave * 32 + lane] = acc;
    __syncthreads();

    if (wave == 0) {
        v8f tot = {};
        #pragma unroll
        for (int w = 0; w < WAVES_PER_BLOCK; ++w)
            tot += red[w * 32 + lane];

        // D layout: lane 0 (N=0) holds rows 0..7 in VGPRs 0..7,
        //           lane 16 (N=0) holds rows 8..15. Column N=0 suffices.
        if ((lane & 15) == 0) {
            const int mbase = hi * 8;
            #pragma unroll
            for (int j = 0; j < 8; ++j) {
                const int row = rowBase + mbase + j;       // row = n*512 + c
                out[row] = tot[j] * mctr[row & (CHANNEL - 1)];
            }
        }
    }
}

extern "C" void kernel_launch(void* const* d_in, const int* in_sizes, int n_in,
                              void* d_out, int out_size, void* d_ws, size_t ws_size,
                              hipStream_t stream) {
    const float* x    = (const float*)d_in[0];   // [32, 512, 56, 56] f32
    const float* mask = (const float*)d_in[1];   // [512, 56, 56] f32
    float* out  = (float*)d_out;                 // [32, 512] f32
    float* mctr = (float*)d_ws;                  // 512 floats of scratch

    gather_mask_center<<<2, 256, 0, stream>>>(mask, mctr);
    fft_mask_rowsum<<<ROWS / 16, 256, 0, stream>>>(x, mctr, out);
}